// SpikeFP64ExtractMod4_43860206027303
// MI455X (gfx1250) — compile-verified
//
#include <hip/hip_runtime.h>
#include <stdint.h>

typedef uint32_t u32;
typedef uint64_t u64;
typedef __attribute__((ext_vector_type(4))) u32 v4u;
typedef __attribute__((ext_vector_type(4))) int  v4i;
typedef __attribute__((ext_vector_type(8))) int  v8i;

#define TILE_ROWS 256
#define ROW_ELTS  64
#define TILE_COLS 14   // columns 0..13 cover all needed fields (0,1,11,12,13); 56B, mult of 4

#if defined(__HIP_DEVICE_COMPILE__) && __has_builtin(__builtin_amdgcn_tensor_load_to_lds)
#define ATHENA_TDM 1
#else
#define ATHENA_TDM 0
#endif

// Exact mux arithmetic of the reference:
//   mux(s,a,b) = s*a + (1-s)*b
//   b0_high = mux(e0, m50, m51); b1_high = mux(e0, m51, 1)
//   b0 = mux(e10, b0_high, e0); b1 = e10*b1_high
//   result[...,0]=b1, result[...,1]=b0 ; second output = sign
__device__ __forceinline__ void compute_store(float sign, float e10, float e0,
                                              float m51, float m50,
                                              float* __restrict__ out,
                                              u64 row, u64 rows) {
  float one_m_e0  = 1.0f - e0;
  float one_m_e10 = 1.0f - e10;
  float b0_high = e0 * m50 + one_m_e0 * m51;
  float b1_high = e0 * m51 + one_m_e0 * 1.0f;
  float b0 = e10 * b0_high + one_m_e10 * e0;
  float b1 = e10 * b1_high;
  float2 r;
  r.x = b1;
  r.y = b0;
  *reinterpret_cast<float2*>(out + row * 2u) = r;   // coalesced b64 store
  out[rows * 2u + row] = sign;                       // coalesced b32 store
}

__global__ __launch_bounds__(TILE_ROWS) void spike_extract_kernel(
    const float* __restrict__ x, float* __restrict__ out, int rows) {
  const int t     = threadIdx.x;
  const int nBase = blockIdx.x * TILE_ROWS;

#if ATHENA_TDM
  __shared__ __align__(16) float sh[TILE_ROWS * TILE_COLS];

  int tileRows = rows - nBase;
  if (tileRows > TILE_ROWS) tileRows = TILE_ROWS;

  if (t < 32) {  // wave 0 only (wave-uniform branch; EXEC stays all-ones in wave 0)
    // ---- Tensor DMA descriptor (D#), per CDNA5 ISA ----
    const u64 gaddr = (u64)(uintptr_t)x + (u64)nBase * (ROW_ELTS * 4u);
    const u32 lds   = (u32)(uintptr_t)(&sh[0]);   // flat->LDS: low 32 bits = LDS byte offset

    // Group 0 (128b): count=1 | lds_addr | global_addr[56:0] | type=2
    v4u g0;
    g0.x = 1u;                                            // count=1, user mode, no gather
    g0.y = lds;                                           // lds_addr (bytes)
    g0.z = (u32)gaddr;                                    // global_addr[31:0]
    g0.w = ((u32)(gaddr >> 32) & 0x01FFFFFFu) | (2u << 30); // addr[56:32] | type=2

    // Group 1 (256b): mask=0, data_size=4B, dims/tile/strides
    const u32 d0 = (u32)ROW_ELTS;        // tensor_dim0 = 64 elements
    const u32 d1 = (u32)tileRows;        // tensor_dim1 = rows in this tile
    const u32 s0 = (u32)ROW_ELTS;        // tensor_dim0_stride = 64 elements
    v8i g1;
    g1[0] = (int)(2u << 16);                               // [17:16] data_size=2 (4B)
    g1[1] = (int)(d0 << 16);                               // [63:48]  tensor_dim0 lo16
    g1[2] = (int)((d0 >> 16) | (d1 << 16));                // dim0 hi16 | dim1 lo16
    g1[3] = (int)((d1 >> 16) | ((u32)TILE_COLS << 16));    // dim1 hi16 | tile_dim0=14
    g1[4] = (int)(d1 & 0xFFFFu);                           // tile_dim1 | tile_dim2=0
    g1[5] = (int)s0;                                       // tensor_dim0_stride lo32
    g1[6] = 0;                                             // stride hi | dim1_stride lo
    g1[7] = 0;                                             // dim1_stride hi

    v4i gz = {0, 0, 0, 0};
#if __has_include(<hip/amd_detail/amd_gfx1250_TDM.h>)
    v8i gz8 = {0, 0, 0, 0, 0, 0, 0, 0};
    __builtin_amdgcn_tensor_load_to_lds(g0, g1, gz, gz, gz8, 0);   // 6-arg toolchain
#else
    __builtin_amdgcn_tensor_load_to_lds(g0, g1, gz, gz, 0);        // 5-arg (ROCm 7.2)
#endif

#if __has_builtin(__builtin_amdgcn_s_wait_tensorcnt)
    __builtin_amdgcn_s_wait_tensorcnt(0);
#else
    asm volatile("s_wait_tensorcnt 0x0" ::: "memory");
#endif
  }
  __syncthreads();

  const int row = nBase + t;
  if (row < rows) {
    // LDS row stride = 14 dwords: 14*t mod 64 is injective over a wave32 -> conflict-free
    const float* r = &sh[t * TILE_COLS];
    compute_store(r[0], r[1], r[11], r[12], r[13], out, (u64)row, (u64)rows);
  }

#else  // Fallback: direct minimal-footprint global loads (same HBM traffic)
  const int row = nBase + t;
  if (row < rows) {
    const float* p = x + (u64)row * ROW_ELTS;
    float2 a = *reinterpret_cast<const float2*>(p);        // sign, e10   (8B aligned)
    float  e0 = p[11];
    float2 m  = *reinterpret_cast<const float2*>(p + 12);  // m51, m50   (8B aligned)
    compute_store(a.x, a.y, e0, m.x, m.y, out, (u64)row, (u64)rows);
  }
#endif
}

extern "C" void kernel_launch(void* const* d_in, const int* in_sizes, int n_in,
                              void* d_out, int out_size, void* d_ws, size_t ws_size,
                              hipStream_t stream) {
  const float* x  = (const float*)d_in[0];
  float*      out = (float*)d_out;
  const int rows   = in_sizes[0] / ROW_ELTS;            // 512*2048 = 1,048,576
  const int blocks = (rows + TILE_ROWS - 1) / TILE_ROWS;
  spike_extract_kernel<<<blocks, TILE_ROWS, 0, stream>>>(x, out, rows);
}